// Model6GCN_72267119722885
// MI455X (gfx1250) — compile-verified
//
#include <hip/hip_runtime.h>
#include <math.h>

#define EPS 1e-6f
#define KN 8          // neighbors per hop
#define D64 64        // feature dim

typedef __attribute__((ext_vector_type(2))) float v2f;
typedef __attribute__((ext_vector_type(8))) float v8f;

__device__ __constant__ int USER_OFF_D[7] = {0, 11577, 11588, 11599, 11610, 11621, 11642};
__device__ __constant__ int URL_OFF_D[5]  = {0, 4733, 4754, 4761, 4772};

__device__ __forceinline__ float wave_sum(float v) {
#pragma unroll
  for (int m = 16; m > 0; m >>= 1) v += __shfl_xor(v, m, 32);
  return v;
}

// ---------------------------------------------------------------------------
// Kernel 1: center embeddings  (ori_user [N,64], url_embed [N,64])
// ---------------------------------------------------------------------------
__global__ void k_center(const int* __restrict__ user_f, const int* __restrict__ url_f,
                         const float* __restrict__ user_tab, const float* __restrict__ url_tab,
                         float* __restrict__ ORI, float* __restrict__ URL, int N) {
  int t = blockIdx.x * blockDim.x + threadIdx.x;
  if (t >= N * D64) return;
  int n = t >> 6, d = t & 63;
  float s = 0.f;
#pragma unroll
  for (int f = 0; f < 7; ++f)
    s += user_tab[(size_t)(user_f[n * 7 + f] + USER_OFF_D[f]) * D64 + d];
  ORI[t] = s;
  s = 0.f;
#pragma unroll
  for (int f = 0; f < 5; ++f)
    s += url_tab[(size_t)(url_f[n * 5 + f] + URL_OFF_D[f]) * D64 + d];
  URL[t] = s;
}

// ---------------------------------------------------------------------------
// Kernel 2: hop-1 embeddings into cols [0,64) of X2 [2*NK, 128]
//   rows [0,NK)   = ori_user_1_url  (url table, 5 feats)
//   rows [NK,2NK) = ori_user_1_user (user table, 7 feats)
// ---------------------------------------------------------------------------
__global__ void k_hop1_embed(const int* __restrict__ u1url_ids, const int* __restrict__ u1user_ids,
                             const float* __restrict__ url_tab, const float* __restrict__ user_tab,
                             float* __restrict__ X2, int NK) {
  int t = blockIdx.x * blockDim.x + threadIdx.x;
  if (t >= 2 * NK * D64) return;
  int e = t >> 6, d = t & 63;
  int b = (e >= NK);
  int nk = b ? (e - NK) : e;
  float s = 0.f;
  if (b == 0) {
#pragma unroll
    for (int f = 0; f < 5; ++f)
      s += url_tab[(size_t)(u1url_ids[nk * 5 + f] + URL_OFF_D[f]) * D64 + d];
  } else {
#pragma unroll
    for (int f = 0; f < 7; ++f)
      s += user_tab[(size_t)(u1user_ids[nk * 7 + f] + USER_OFF_D[f]) * D64 + d];
  }
  X2[(size_t)e * 128 + d] = s;
}

// ---------------------------------------------------------------------------
// Hop-2 attention helper: embed 8 neighbors, cosine vs center, softmax,
// relu(e*w)*p accumulation. One wave; lane owns dims (lane, lane+32).
// ---------------------------------------------------------------------------
template <int F>
__device__ void attn_accum(const int* __restrict__ ids, const float* __restrict__ wgt,
                           const float* __restrict__ tab, const int* offs,
                           int nk, int lane, float c0, float c1, float ncl,
                           float& acc0, float& acc1) {
  float e0[KN], e1[KN], p[KN];
#pragma unroll
  for (int j = 0; j < KN; ++j) {
    const int* idp = ids + ((size_t)nk * KN + j) * F;
    float s0 = 0.f, s1 = 0.f;
#pragma unroll
    for (int f = 0; f < F; ++f) {
      const float* row = tab + (size_t)(idp[f] + offs[f]) * D64;
      s0 += row[lane];
      s1 += row[lane + 32];
    }
    e0[j] = s0; e1[j] = s1;
    float dot = wave_sum(s0 * c0 + s1 * c1);
    float ne  = wave_sum(s0 * s0 + s1 * s1);
    p[j] = dot / (fmaxf(sqrtf(ne), EPS) * ncl);
  }
  float mx = p[0];
#pragma unroll
  for (int j = 1; j < KN; ++j) mx = fmaxf(mx, p[j]);
  float sum = 0.f;
#pragma unroll
  for (int j = 0; j < KN; ++j) { p[j] = __expf(p[j] - mx); sum += p[j]; }
  float inv = 1.f / sum;
#pragma unroll
  for (int j = 0; j < KN; ++j) {
    float wj = wgt[(size_t)nk * KN + j];
    float pj = p[j] * inv;
    acc0 += fmaxf(e0[j] * wj, 0.f) * pj;
    acc1 += fmaxf(e1[j] * wj, 0.f) * pj;
  }
}

// ---------------------------------------------------------------------------
// Kernel 3: hop-2 attention -> cols [64,128) of X2. One wave per (branch,n,k).
// ---------------------------------------------------------------------------
__global__ void k_hop2_attn(const int* __restrict__ u2u_ids, const float* __restrict__ u2u_w,
                            const int* __restrict__ u2url_ids, const float* __restrict__ u2url_w,
                            const int* __restrict__ url2u_ids, const float* __restrict__ url2u_w,
                            const int* __restrict__ url2url_ids, const float* __restrict__ url2url_w,
                            const float* __restrict__ user_tab, const float* __restrict__ url_tab,
                            float* __restrict__ X2, int NK) {
  int w = (blockIdx.x * blockDim.x + threadIdx.x) >> 5;
  int lane = threadIdx.x & 31;
  if (w >= 2 * NK) return;
  int b = (w >= NK);
  int nk = b ? (w - NK) : w;
  float* rowp = X2 + (size_t)w * 128;
  float c0 = rowp[lane], c1 = rowp[lane + 32];
  float ncl = fmaxf(sqrtf(wave_sum(c0 * c0 + c1 * c1)), EPS);
  float acc0 = 0.f, acc1 = 0.f;
  if (b == 0) {  // url branch: url_2_user (user table) + url_2_url (url table)
    attn_accum<7>(url2u_ids, url2u_w, user_tab, USER_OFF_D, nk, lane, c0, c1, ncl, acc0, acc1);
    attn_accum<5>(url2url_ids, url2url_w, url_tab, URL_OFF_D, nk, lane, c0, c1, ncl, acc0, acc1);
  } else {       // user branch: user_2_user + user_2_url
    attn_accum<7>(u2u_ids, u2u_w, user_tab, USER_OFF_D, nk, lane, c0, c1, ncl, acc0, acc1);
    attn_accum<5>(u2url_ids, u2url_w, url_tab, URL_OFF_D, nk, lane, c0, c1, ncl, acc0, acc1);
  }
  rowp[64 + lane] = 0.5f * acc0;
  rowp[96 + lane] = 0.5f * acc1;
}

// ---------------------------------------------------------------------------
// Kernel 4/6: Y[M,64] = relu(X[M,128] @ W[64,128]^T + b) via V_WMMA_F32_16X16X4_F32.
// Block = 256 threads = 8 waves; each wave owns a 16-row tile, 4 N-tiles of 16.
// W^T staged in LDS as interleaved k-pairs -> one ds_load_b64 per B fragment.
// ---------------------------------------------------------------------------
__global__ __launch_bounds__(256) void k_gemm_relu_128x64(
    const float* __restrict__ X, const float* __restrict__ W,
    const float* __restrict__ bias, float* __restrict__ Y, int M) {
  __shared__ float2 WtP[64 * 64];  // WtP[p*64+n] = (W[n][2p], W[n][2p+1]), 32 KB
  __shared__ float bsh[64];
  int tid = threadIdx.x;
  for (int i = tid; i < 4096; i += 256) {
    int p = i >> 6, n = i & 63;
    const float* wp = W + n * 128 + 2 * p;
    WtP[i] = make_float2(wp[0], wp[1]);
  }
  if (tid < 64) bsh[tid] = bias[tid];
  __syncthreads();

  int wave = tid >> 5, lane = tid & 31;
  int row0 = (blockIdx.x * 8 + wave) * 16;
  if (row0 >= M) return;

  // A fragment layout (f32 16x4): lanes 0-15 -> M=lane, K=kk..kk+1;
  //                               lanes 16-31 -> M=lane-16, K=kk+2..kk+3.
  int half = lane >> 4;          // 0 or 1
  int nlo = lane & 15;
  int arow = row0 + nlo;
  const float* xr = X + (size_t)arow * 128 + half * 2;
  __builtin_prefetch(xr, 0, 1);  // global_prefetch_b8: pull A tile toward L0/L2

  v8f acc[4] = {};
  for (int t = 0; t < 32; ++t) {         // kk = 4*t
    v2f a = *reinterpret_cast<const v2f*>(xr + 4 * t);
    int p = 2 * t + half;
#pragma unroll
    for (int nt = 0; nt < 4; ++nt) {
      float2 wv = WtP[p * 64 + nt * 16 + nlo];
      v2f bf; bf[0] = wv.x; bf[1] = wv.y;
      acc[nt] = __builtin_amdgcn_wmma_f32_16x16x4_f32(
          false, a, false, bf, (short)0, acc[nt], false, false);
    }
  }

  // D layout: lanes 0-15 hold rows M=0..7 (vgpr r), lanes 16-31 rows M=8..15.
  int rbase = row0 + half * 8;
#pragma unroll
  for (int nt = 0; nt < 4; ++nt) {
    int col = nt * 16 + nlo;
    float bb = bsh[col];
#pragma unroll
    for (int r = 0; r < 8; ++r)
      Y[(size_t)(rbase + r) * D64 + col] = fmaxf(acc[nt][r] + bb, 0.f);
  }
}

// ---------------------------------------------------------------------------
// Kernel 5: hop-1 attention from H [2NK,64] -> X1 [N,128]. One wave per n.
// ---------------------------------------------------------------------------
__global__ void k_hop1_attn(const float* __restrict__ H,
                            const float* __restrict__ w_url, const float* __restrict__ w_user,
                            const float* __restrict__ ORI, float* __restrict__ X1,
                            int N, int NK) {
  int w = (blockIdx.x * blockDim.x + threadIdx.x) >> 5;
  int lane = threadIdx.x & 31;
  if (w >= N) return;
  float c0 = ORI[(size_t)w * D64 + lane];
  float c1 = ORI[(size_t)w * D64 + 32 + lane];
  float ncl = fmaxf(sqrtf(wave_sum(c0 * c0 + c1 * c1)), EPS);
  float acc0 = 0.f, acc1 = 0.f;
#pragma unroll
  for (int s = 0; s < 2; ++s) {
    const float* Hb = s ? (H + (size_t)NK * D64) : H;   // s=0: user_1_url rows, s=1: user_1_user rows
    const float* wg = s ? w_user : w_url;
    float e0[KN], e1[KN], p[KN];
#pragma unroll
    for (int j = 0; j < KN; ++j) {
      const float* hr = Hb + ((size_t)w * KN + j) * D64;
      float s0 = hr[lane], s1 = hr[lane + 32];
      e0[j] = s0; e1[j] = s1;
      float dot = wave_sum(s0 * c0 + s1 * c1);
      float ne  = wave_sum(s0 * s0 + s1 * s1);
      p[j] = dot / (fmaxf(sqrtf(ne), EPS) * ncl);
    }
    float mx = p[0];
#pragma unroll
    for (int j = 1; j < KN; ++j) mx = fmaxf(mx, p[j]);
    float sum = 0.f;
#pragma unroll
    for (int j = 0; j < KN; ++j) { p[j] = __expf(p[j] - mx); sum += p[j]; }
    float inv = 1.f / sum;
#pragma unroll
    for (int j = 0; j < KN; ++j) {
      float wj = wg[(size_t)w * KN + j];
      float pj = p[j] * inv;
      acc0 += fmaxf(e0[j] * wj, 0.f) * pj;
      acc1 += fmaxf(e1[j] * wj, 0.f) * pj;
    }
  }
  float* xr = X1 + (size_t)w * 128;
  xr[lane] = c0;
  xr[32 + lane] = c1;
  xr[64 + lane] = 0.5f * acc0;
  xr[96 + lane] = 0.5f * acc1;
}

// ---------------------------------------------------------------------------
// Kernel 7: logits = concat(user_embed, url_embed) @ Wo^T + bo; softmax(2).
// One wave per n.
// ---------------------------------------------------------------------------
__global__ void k_final(const float* __restrict__ E1, const float* __restrict__ URL,
                        const float* __restrict__ Wo, const float* __restrict__ bo,
                        float* __restrict__ out, int N) {
  int w = (blockIdx.x * blockDim.x + threadIdx.x) >> 5;
  int lane = threadIdx.x & 31;
  if (w >= N) return;
  float ue0 = E1[(size_t)w * D64 + lane], ue1 = E1[(size_t)w * D64 + 32 + lane];
  float ur0 = URL[(size_t)w * D64 + lane], ur1 = URL[(size_t)w * D64 + 32 + lane];
  float l0 = wave_sum(ue0 * Wo[lane] + ue1 * Wo[32 + lane] +
                      ur0 * Wo[64 + lane] + ur1 * Wo[96 + lane]);
  float l1 = wave_sum(ue0 * Wo[128 + lane] + ue1 * Wo[160 + lane] +
                      ur0 * Wo[192 + lane] + ur1 * Wo[224 + lane]);
  if (lane == 0) {
    l0 += bo[0]; l1 += bo[1];
    float m = fmaxf(l0, l1);
    float a = __expf(l0 - m), b = __expf(l1 - m);
    float inv = 1.f / (a + b);
    out[(size_t)w * 2]     = a * inv;
    out[(size_t)w * 2 + 1] = b * inv;
  }
}

// ---------------------------------------------------------------------------
extern "C" void kernel_launch(void* const* d_in, const int* in_sizes, int n_in,
                              void* d_out, int out_size, void* d_ws, size_t ws_size,
                              hipStream_t stream) {
  (void)n_in; (void)out_size; (void)ws_size;
  const int* user_f_list  = (const int*)d_in[1];
  const int* url_f_list   = (const int*)d_in[2];
  const int* u1user_ids   = (const int*)d_in[3];
  const float* u1user_w   = (const float*)d_in[4];
  const int* u1url_ids    = (const int*)d_in[5];
  const float* u1url_w    = (const float*)d_in[6];
  const int* u2u_ids      = (const int*)d_in[7];
  const float* u2u_w      = (const float*)d_in[8];
  const int* u2url_ids    = (const int*)d_in[9];
  const float* u2url_w    = (const float*)d_in[10];
  const int* url2u_ids    = (const int*)d_in[11];
  const float* url2u_w    = (const float*)d_in[12];
  const int* url2url_ids  = (const int*)d_in[13];
  const float* url2url_w  = (const float*)d_in[14];
  const float* user_tab   = (const float*)d_in[15];
  const float* url_tab    = (const float*)d_in[16];
  const float* W2 = (const float*)d_in[17];
  const float* b2 = (const float*)d_in[18];
  const float* W1 = (const float*)d_in[19];
  const float* b1 = (const float*)d_in[20];
  const float* Wo = (const float*)d_in[21];
  const float* bo = (const float*)d_in[22];

  const int N  = in_sizes[1] / 7;   // 8192
  const int NK = N * KN;            // 65536

  // Workspace layout (floats)
  float* ws  = (float*)d_ws;
  float* X2  = ws;                                  // [2*NK, 128]
  float* H   = X2 + (size_t)2 * NK * 128;           // [2*NK, 64]
  float* ORI = H  + (size_t)2 * NK * D64;           // [N, 64]
  float* URL = ORI + (size_t)N * D64;               // [N, 64]
  float* X1  = URL + (size_t)N * D64;               // [N, 128]
  float* E1  = X1 + (size_t)N * 128;                // [N, 64]

  const int B = 256;

  k_center<<<(N * D64 + B - 1) / B, B, 0, stream>>>(
      user_f_list, url_f_list, user_tab, url_tab, ORI, URL, N);

  k_hop1_embed<<<(2 * NK * D64 + B - 1) / B, B, 0, stream>>>(
      u1url_ids, u1user_ids, url_tab, user_tab, X2, NK);

  k_hop2_attn<<<(2 * NK * 32 + B - 1) / B, B, 0, stream>>>(
      u2u_ids, u2u_w, u2url_ids, u2url_w,
      url2u_ids, url2u_w, url2url_ids, url2url_w,
      user_tab, url_tab, X2, NK);

  k_gemm_relu_128x64<<<(2 * NK + 127) / 128, B, 0, stream>>>(X2, W2, b2, H, 2 * NK);

  k_hop1_attn<<<(N * 32 + B - 1) / B, B, 0, stream>>>(
      H, u1url_w, u1user_w, ORI, X1, N, NK);

  k_gemm_relu_128x64<<<(N + 127) / 128, B, 0, stream>>>(X1, W1, b1, E1, N);

  k_final<<<(N * 32 + B - 1) / B, B, 0, stream>>>(E1, URL, Wo, bo, (float*)d_out, N);
}